// GraphConvolution_26834955665849
// MI455X (gfx1250) — compile-verified
//
#include <hip/hip_runtime.h>

typedef float v2f __attribute__((ext_vector_type(2)));
typedef float v8f __attribute__((ext_vector_type(8)));

#define NNODES 50000
#define NB 4
#define FDIM 64
#define NEDGE 800000
#define NROWS (NB * NNODES)   // 200000 flattened rows
#define NRB   (NROWS / 16)    // 12500 row blocks (exact)

// ---------------------------------------------------------------------------
// Dense GEMM: pre1 = X @ W1, pre2 = X @ W2   (X: [NROWS,64] fp32, exact fp32 WMMA)
// One wave per 16-row block; K=64 chained as 16x V_WMMA_F32_16X16X4_F32.
// ---------------------------------------------------------------------------
__global__ __launch_bounds__(256) void gcn_gemm_f32wmma(
    const float* __restrict__ x, const float* __restrict__ W1,
    const float* __restrict__ W2, float* __restrict__ pre1,
    float* __restrict__ pre2) {
  const int wave = (blockIdx.x * 256 + threadIdx.x) >> 5;
  if (wave >= NRB) return;                 // wave-uniform: EXEC stays all-ones
  const int lane = threadIdx.x & 31;
  const int half = lane >> 4;              // 0: K even pair, 1: K odd pair
  const int lr   = lane & 15;

  const float* xrow = x + (size_t)(wave * 16 + lr) * FDIM + 2 * half;

  for (int wsel = 0; wsel < 2; ++wsel) {
    const float* W  = wsel ? W2 : W1;
    float*      pre = wsel ? pre2 : pre1;

    v8f acc0 = {}, acc1 = {}, acc2 = {}, acc3 = {};
#pragma unroll
    for (int s = 0; s < 16; ++s) {
      // A fragment (16x4): lane<16 holds K=4s,4s+1 ; lane>=16 holds K=4s+2,4s+3
      v2f a;
      a.x = xrow[4 * s + 0];
      a.y = xrow[4 * s + 1];
      // B fragments (4x16): VGPR0=K row (4s+2*half), VGPR1=next K row, N=lr
      const float* wk = W + (size_t)(4 * s + 2 * half) * FDIM + lr;
      v2f b0, b1, b2, b3;
      b0.x = wk[0];  b0.y = wk[FDIM];
      b1.x = wk[16]; b1.y = wk[FDIM + 16];
      b2.x = wk[32]; b2.y = wk[FDIM + 32];
      b3.x = wk[48]; b3.y = wk[FDIM + 48];
      acc0 = __builtin_amdgcn_wmma_f32_16x16x4_f32(false, a, false, b0, (short)0, acc0, false, false);
      acc1 = __builtin_amdgcn_wmma_f32_16x16x4_f32(false, a, false, b1, (short)0, acc1, false, false);
      acc2 = __builtin_amdgcn_wmma_f32_16x16x4_f32(false, a, false, b2, (short)0, acc2, false, false);
      acc3 = __builtin_amdgcn_wmma_f32_16x16x4_f32(false, a, false, b3, (short)0, acc3, false, false);
    }
    // C/D layout: VGPR r -> row M = r + 8*half, column N = lr
    float* prow = pre + (size_t)(wave * 16 + 8 * half) * FDIM + lr;
#pragma unroll
    for (int r = 0; r < 8; ++r) {
      prow[(size_t)r * FDIM +  0] = acc0[r];
      prow[(size_t)r * FDIM + 16] = acc1[r];
      prow[(size_t)r * FDIM + 32] = acc2[r];
      prow[(size_t)r * FDIM + 48] = acc3[r];
    }
  }
}

// ---------------------------------------------------------------------------
// Non-returning relaxed agent-scope float add -> native global_atomic_add_f32
// (no pre-op value requested => atomic-without-return, STOREcnt-tracked).
// ---------------------------------------------------------------------------
__device__ __forceinline__ void atomic_fadd_noret(float* p, float v) {
  (void)__hip_atomic_fetch_add(p, v, __ATOMIC_RELAXED, __HIP_MEMORY_SCOPE_AGENT);
}

// ---------------------------------------------------------------------------
// SpMM scatter: out[b][rows[e]][:] += vals[e] * pre[b][cols[e]][:]
// One wave per edge; lane owns features [2*lane, 2*lane+1] (coalesced float2).
// pre/out are L2-resident (121 MB working set < 192 MB L2).
// ---------------------------------------------------------------------------
__global__ __launch_bounds__(256) void gcn_spmm(
    const int* __restrict__ rows, const int* __restrict__ cols,
    const float* __restrict__ vals, const float* __restrict__ pre,
    float* __restrict__ out) {
  const int e = blockIdx.x * 8 + (threadIdx.x >> 5);
  if (e >= NEDGE) return;
  const int lane = threadIdx.x & 31;
  const int r = rows[e];
  const int c = cols[e];
  const float v = vals[e];
  const float2* src = (const float2*)(pre + (size_t)c * FDIM) + lane;
  float* dst = out + (size_t)r * FDIM + 2 * lane;
#pragma unroll
  for (int b = 0; b < NB; ++b) {
    float2 p = src[(size_t)b * (NNODES * FDIM / 2)];
    atomic_fadd_noret(dst + (size_t)b * NNODES * FDIM + 0, v * p.x);
    atomic_fadd_noret(dst + (size_t)b * NNODES * FDIM + 1, v * p.y);
  }
}

// ---------------------------------------------------------------------------
// Zero-init of out (harness poisons d_out; atomics need a clean accumulator).
// ---------------------------------------------------------------------------
__global__ __launch_bounds__(256) void gcn_zero(float4* __restrict__ out, int n4) {
  int i = blockIdx.x * 256 + threadIdx.x;
  if (i < n4) out[i] = make_float4(0.f, 0.f, 0.f, 0.f);
}

// ---------------------------------------------------------------------------
// Fused bias + ReLU over [NROWS, 64] as float4 (16 float4 per row).
// ---------------------------------------------------------------------------
__global__ __launch_bounds__(256) void gcn_bias_relu(
    float4* __restrict__ out, const float4* __restrict__ bias4, int n4) {
  int i = blockIdx.x * 256 + threadIdx.x;
  if (i >= n4) return;
  float4 o = out[i];
  float4 bb = bias4[i & 15];   // FDIM/4 == 16 float4 per row
  o.x = fmaxf(o.x + bb.x, 0.f);
  o.y = fmaxf(o.y + bb.y, 0.f);
  o.z = fmaxf(o.z + bb.z, 0.f);
  o.w = fmaxf(o.w + bb.w, 0.f);
  out[i] = o;
}

extern "C" void kernel_launch(void* const* d_in, const int* in_sizes, int n_in,
                              void* d_out, int out_size, void* d_ws, size_t ws_size,
                              hipStream_t stream) {
  (void)in_sizes; (void)n_in; (void)out_size; (void)ws_size;
  const float* x     = (const float*)d_in[0];
  const int*   rows1 = (const int*)d_in[1];
  const int*   cols1 = (const int*)d_in[2];
  const float* vals1 = (const float*)d_in[3];
  const int*   rows2 = (const int*)d_in[4];
  const int*   cols2 = (const int*)d_in[5];
  const float* vals2 = (const float*)d_in[6];
  const float* W1    = (const float*)d_in[7];
  const float* W2    = (const float*)d_in[8];
  const float* bias  = (const float*)d_in[9];
  float* out  = (float*)d_out;
  float* pre1 = (float*)d_ws;                       // [NROWS,64]
  float* pre2 = pre1 + (size_t)NROWS * FDIM;        // [NROWS,64]

  const int n4 = NROWS * FDIM / 4;                  // 3,200,000

  gcn_zero<<<(n4 + 255) / 256, 256, 0, stream>>>((float4*)out, n4);
  gcn_gemm_f32wmma<<<(NRB * 32 + 255) / 256, 256, 0, stream>>>(x, W1, W2, pre1, pre2);
  gcn_spmm<<<(NEDGE + 7) / 8, 256, 0, stream>>>(rows1, cols1, vals1, pre1, out);
  gcn_spmm<<<(NEDGE + 7) / 8, 256, 0, stream>>>(rows2, cols2, vals2, pre2, out);
  gcn_bias_relu<<<(n4 + 255) / 256, 256, 0, stream>>>((float4*)out, (const float4*)bias, n4);
}